// FrankenMACE_78761110274484
// MI455X (gfx1250) — compile-verified
//
#include <hip/hip_runtime.h>
#include <math.h>

// MACE-like 2-layer equivariant GNN for gfx1250 (MI455X), fp32 WMMA path.
// featsT[n][L=16][k=32] layout; channel-mix linears, the per-edge CG
// contraction AND the per-node symmetric (quadratic) contraction all run on
// V_WMMA_F32_16X16X4_F32. Weights (32x32 and the flattened 256x16 Wcg2) are
// pair-swizzled (row-pair, col, 2) so every WMMA A/B fragment from LDS or
// global is a single aligned b64 load.

#define ZDIM    10
#define KCH     32
#define LDIM    16
#define NBESSEL 8
#define NLAYER  2

typedef __attribute__((ext_vector_type(2))) float v2f;
typedef __attribute__((ext_vector_type(8))) float v8f;

__device__ __forceinline__ v8f wmma4(v2f a, v2f b, v8f c) {
  // D(16x16 f32) += A(16x4 f32) * B(4x16 f32)
  return __builtin_amdgcn_wmma_f32_16x16x4_f32(
      /*neg_a=*/false, a, /*neg_b=*/false, b,
      /*c_mod=*/(short)0, c, /*reuse_a=*/false, /*reuse_b=*/false);
}

// ---------------------------------------------------------------- geometry
__global__ void geom_kernel(const float* __restrict__ pos,
                            const float* __restrict__ shifts,
                            const int* __restrict__ eidx,
                            float* __restrict__ Y, float* __restrict__ EF,
                            int E) {
  int e = blockIdx.x * blockDim.x + threadIdx.x;
  if (e >= E) return;
  int snd = eidx[e], rcv = eidx[E + e];
  float vx = pos[rcv * 3 + 0] - pos[snd * 3 + 0] + shifts[e * 3 + 0];
  float vy = pos[rcv * 3 + 1] - pos[snd * 3 + 1] + shifts[e * 3 + 1];
  float vz = pos[rcv * 3 + 2] - pos[snd * 3 + 2] + shifts[e * 3 + 2];
  float r = sqrtf(fmaxf(vx * vx + vy * vy + vz * vz, 1e-12f));
  float inv = 1.0f / r;
  float x = vx * inv, y = vy * inv, z = vz * inv;

  const float S3 = 1.7320508075688772f, S5 = 2.2360679774997896f;
  const float S15 = 3.8729833462074170f, S35_8 = 2.0916500663351889f;
  const float S105 = 10.246950765959598f, S21_8 = 1.6201851746019651f;
  const float S7 = 2.6457513110645907f;
  float* Ye = Y + (size_t)e * 16;
  Ye[0] = 1.0f;
  Ye[1] = S3 * x;  Ye[2] = S3 * y;  Ye[3] = S3 * z;
  Ye[4] = S15 * x * y;  Ye[5] = S15 * y * z;
  Ye[6] = 0.5f * S5 * (3.0f * z * z - 1.0f);
  Ye[7] = S15 * x * z;  Ye[8] = 0.5f * S15 * (x * x - y * y);
  Ye[9]  = S35_8 * y * (3.0f * x * x - y * y);
  Ye[10] = S105 * x * y * z;
  Ye[11] = S21_8 * y * (5.0f * z * z - 1.0f);
  Ye[12] = 0.5f * S7 * (5.0f * z * z * z - 3.0f * z);
  Ye[13] = S21_8 * x * (5.0f * z * z - 1.0f);
  Ye[14] = 0.5f * S105 * z * (x * x - y * y);
  Ye[15] = S35_8 * x * (x * x - 3.0f * y * y);

  float uu = fminf(fmaxf(r * 0.2f, 0.0f), 1.0f);
  float u2 = uu * uu, u4 = u2 * u2, u6 = u4 * u2, u7 = u6 * uu, u8 = u7 * uu;
  float f = 1.0f - 28.0f * u6 + 48.0f * u7 - 21.0f * u8;
  float pir = 3.14159265358979323846f * r * 0.2f;
  float c = 0.6324555320336759f * inv * f;  // sqrt(2/5)/r * cutoff
  float* efe = EF + (size_t)e * 8;
#pragma unroll
  for (int n = 1; n <= 8; ++n) efe[n - 1] = c * __sinf((float)n * pir);
}

// ------------------------------------------------------- embedding + species
__global__ void embed_kernel(const float* __restrict__ attrs,
                             const float* __restrict__ Wemb,
                             float* __restrict__ feats, int* __restrict__ spec,
                             int N) {
  int t = blockIdx.x * blockDim.x + threadIdx.x;
  if (t >= N * KCH) return;
  int n = t >> 5, k = t & 31;
  float s = 0.0f;
#pragma unroll
  for (int z = 0; z < ZDIM; ++z) s += attrs[n * ZDIM + z] * Wemb[z * KCH + k];
  float* f = feats + (size_t)n * 512;
  f[k] = s;  // L=0 row
#pragma unroll
  for (int l = 1; l < LDIM; ++l) f[l * 32 + k] = 0.0f;
  if (k == 0) {
    int zb = 0;
    float best = attrs[n * ZDIM];
    for (int z = 1; z < ZDIM; ++z) {
      float v = attrs[n * ZDIM + z];
      if (v > best) { best = v; zb = z; }
    }
    spec[n] = zb;
  }
}

__global__ void zero_kernel(float* __restrict__ p, int n) {
  int i = blockIdx.x * blockDim.x + threadIdx.x;
  if (i < n) p[i] = 0.0f;
}

// ---------------------------------------------- weight pair-swizzle (32 cols)
// SW[p*64 + c*2 + o] = W[2p+o][c]  (p=row pair, c=col 0..31, o=0/1)
__global__ void swizzle_w_kernel(const float* __restrict__ src,
                                 float* __restrict__ dst, int nmat) {
  int i = blockIdx.x * blockDim.x + threadIdx.x;
  if (i >= nmat * 1024) return;
  int idx = i & 1023;
  int mat = i >> 10;
  int o = idx & 1, c = (idx >> 1) & 31, p = idx >> 6;
  dst[i] = src[mat * 1024 + (2 * p + o) * 32 + c];
}

// ------------------------------------- Wcg2 flat (256x16) pair-swizzle
// SW[p*32 + t*2 + o] = Wflat[2p+o][t]  (Wflat[ab][t] = Wcg2[a][b][t])
__global__ void swizzle_w16_kernel(const float* __restrict__ src,
                                   float* __restrict__ dst) {
  int i = blockIdx.x * blockDim.x + threadIdx.x;
  if (i >= 4096) return;
  int o = i & 1, t = (i >> 1) & 15, p = i >> 5;
  dst[i] = src[(2 * p + o) * 16 + t];
}

// -------------------------------------------------- linear_up (wave per node)
// up_swizzled[n] = featsT[n](16x32) @ W_up(32x32); D row-pairs stored as b64
__global__ void linear_up_kernel(const float* __restrict__ feats,
                                 const float* __restrict__ WupS,  // swizzled
                                 float* __restrict__ up) {
  __shared__ __align__(16) float sW[1024];
  int tid = threadIdx.x;
  for (int i = tid; i < 1024; i += blockDim.x) sW[i] = WupS[i];
  __syncthreads();
  int wv = tid >> 5, lane = tid & 31;
  int n = blockIdx.x * 8 + wv;
  int m = lane & 15, half = lane >> 4, aoff = half * 2;
  const float* fr = feats + (size_t)n * 512 + m * 32;
#pragma unroll
  for (int ct = 0; ct < 2; ++ct) {
    int col = ct * 16 + m;
    v8f acc = {0.f, 0.f, 0.f, 0.f, 0.f, 0.f, 0.f, 0.f};
#pragma unroll
    for (int s = 0; s < 8; ++s) {
      v2f a = *(const v2f*)(fr + 4 * s + aoff);
      v2f b = *(const v2f*)(&sW[(2 * s + half) * 64 + col * 2]);
      acc = wmma4(a, b, acc);
    }
    // store row pairs (2j,2j+1) -> pair index j + half*4, b64 each
    float* ob = up + (size_t)n * 512 + half * 256 + col * 2;
#pragma unroll
    for (int j = 0; j < 4; ++j) {
      v2f v; v.x = acc[2 * j]; v.y = acc[2 * j + 1];
      *(v2f*)(ob + j * 64) = v;
    }
  }
}

// ------------------------------------------ edge message kernel (wave/edge)
// Rw = silu(EF @ Wr1) @ Wr2 ; C[a,t] = sum_s Y[s]*Wcg[a,s,t]
// msgT(16x32) = C^T(16x16) @ (up_sw[snd] * Rw)(16x32); atomic scatter to agg
__global__ void edge_kernel(const float* __restrict__ Y,
                            const float* __restrict__ EF,
                            const float* __restrict__ Wr1,   // [8,32] linear
                            const float* __restrict__ Wr2,   // [32,32] linear
                            const float* __restrict__ Wcg,   // [16,16,16]
                            const float* __restrict__ up,    // swizzled
                            const int* __restrict__ eidx,
                            float* __restrict__ agg, int E) {
  __shared__ __align__(16) float sWcg[4096];
  __shared__ __align__(16) float sWr1[256];
  __shared__ __align__(16) float sWr2[1024];
  __shared__ float sH[8][32];
  __shared__ float sRW[8][32];
  int tid = threadIdx.x;
  for (int i = tid; i < 4096; i += blockDim.x) sWcg[i] = Wcg[i];
  for (int i = tid; i < 256; i += blockDim.x) sWr1[i] = Wr1[i];
  for (int i = tid; i < 1024; i += blockDim.x) sWr2[i] = Wr2[i];
  __syncthreads();

  int wv = tid >> 5, lane = tid & 31;
  int e = blockIdx.x * 8 + wv;
  int snd = eidx[e], rcv = eidx[E + e];

  // prefetch the gathered 2KB up-tile (64B stride across 32 lanes) so it is
  // in flight while we do the radial MLP + CG A-fragment math
  const float* upr = up + (size_t)snd * 512;
  __builtin_prefetch(upr + lane * 16, 0, 3);

  // radial MLP: lane j owns hidden unit j, then channel j
  const float* ef = EF + (size_t)e * 8;
  float h = 0.0f;
#pragma unroll
  for (int b = 0; b < NBESSEL; ++b) h += ef[b] * sWr1[b * 32 + lane];
  h = h / (1.0f + __expf(-h));  // silu
  sH[wv][lane] = h;
  __syncthreads();
  float rw = 0.0f;
#pragma unroll
  for (int j = 0; j < 32; ++j) rw += sH[wv][j] * sWr2[j * 32 + lane];
  sRW[wv][lane] = rw;
  __syncthreads();

  float y[16];
  const float* Ye = Y + (size_t)e * 16;
#pragma unroll
  for (int s = 0; s < 16; ++s) y[s] = Ye[s];

  int m = lane & 15, half = lane >> 4, aoff = half * 2;
  // A-fragments of C^T: lane row t=m, K slots a = 4s+aoff(+1)
  v2f A[4];
#pragma unroll
  for (int s = 0; s < 4; ++s) {
    int a0 = 4 * s + aoff;
    float ax = 0.0f, ay = 0.0f;
#pragma unroll
    for (int q = 0; q < 16; ++q) {
      ax += y[q] * sWcg[(a0 * 16 + q) * 16 + m];
      ay += y[q] * sWcg[((a0 + 1) * 16 + q) * 16 + m];
    }
    A[s].x = ax; A[s].y = ay;
  }

#pragma unroll
  for (int ct = 0; ct < 2; ++ct) {
    int col = ct * 16 + m;
    float rwc = sRW[wv][col];
    v8f acc = {0.f, 0.f, 0.f, 0.f, 0.f, 0.f, 0.f, 0.f};
#pragma unroll
    for (int s = 0; s < 4; ++s) {
      // pair p = (4s+aoff)/2 = 2s+half : one b64 load gives B rows a0,a0+1
      v2f uv = *(const v2f*)(upr + (2 * s + half) * 64 + col * 2);
      v2f b; b.x = uv.x * rwc; b.y = uv.y * rwc;
      acc = wmma4(A[s], b, acc);
    }
    float* ag = agg + (size_t)rcv * 512 + col;
#pragma unroll
    for (int r = 0; r < 8; ++r)
      atomicAdd(ag + (r + half * 8) * 32, acc[r] * 0.0625f);  // /AVG_NEIGH
  }
}

// ------------------------------------- node update (wave/node): lin+quad+skip
// quad as GEMM: quadT(16x32) = Wcg2_flatT(16x256) @ E^T(256x32),
// E^T[ab][k] = mid[k,a]*mid[k,b]; WMMA C-accumulator carries midT so the
// result is (mid+quad)T directly in [L][k] layout for the W_prod GEMM.
__global__ void node_update_kernel(const float* __restrict__ agg,
                                   const float* __restrict__ featsIn,
                                   const float* __restrict__ WlinS,   // swz
                                   const float* __restrict__ WprodS,  // swz
                                   const float* __restrict__ WskipS,  // [Z] swz
                                   const float* __restrict__ Wcg2S,   // 256x16 swz
                                   const int* __restrict__ spec,
                                   float* __restrict__ featsOut,
                                   float* __restrict__ out, int layer) {
  __shared__ __align__(16) float sWlin[1024];
  __shared__ __align__(16) float sWprod[1024];
  __shared__ __align__(16) float sWcg2[4096];
  __shared__ __align__(16) float sMQ[8][512];  // per-wave midT/(mid+quad)T
  int tid = threadIdx.x;
  for (int i = tid; i < 1024; i += blockDim.x) sWlin[i] = WlinS[i];
  for (int i = tid; i < 1024; i += blockDim.x) sWprod[i] = WprodS[i];
  for (int i = tid; i < 4096; i += blockDim.x) sWcg2[i] = Wcg2S[i];
  __syncthreads();

  int wv = tid >> 5, lane = tid & 31;
  int n = blockIdx.x * 8 + wv;
  int m = lane & 15, half = lane >> 4, aoff = half * 2;

  // midT = aggT @ W_lin -> sMQ
  const float* ar = agg + (size_t)n * 512 + m * 32;
#pragma unroll
  for (int ct = 0; ct < 2; ++ct) {
    int col = ct * 16 + m;
    v8f acc = {0.f, 0.f, 0.f, 0.f, 0.f, 0.f, 0.f, 0.f};
#pragma unroll
    for (int s = 0; s < 8; ++s) {
      v2f a = *(const v2f*)(ar + 4 * s + aoff);
      v2f b = *(const v2f*)(&sWlin[(2 * s + half) * 64 + col * 2]);
      acc = wmma4(a, b, acc);
    }
#pragma unroll
    for (int r = 0; r < 8; ++r)
      sMQ[wv][(r + half * 8) * 32 + col] = acc[r];
  }
  __syncthreads();

  // quad GEMM per column tile; each tile touches only its own 16 columns of
  // sMQ, so no cross-tile hazard (same-wave LDS ops are in order).
#pragma unroll
  for (int ct = 0; ct < 2; ++ct) {
    int col = ct * 16 + m;
    float mb[16];
#pragma unroll
    for (int a = 0; a < 16; ++a) mb[a] = sMQ[wv][a * 32 + col];
    // half-dependent b-part registers: rb0[j]=mb[4j+2*half], rb1[j]=mb[4j+1+2*half]
    float rb0[4], rb1[4];
#pragma unroll
    for (int j = 0; j < 4; ++j) {
      rb0[j] = half ? mb[4 * j + 2] : mb[4 * j];
      rb1[j] = half ? mb[4 * j + 3] : mb[4 * j + 1];
    }
    v8f acc;
#pragma unroll
    for (int r = 0; r < 8; ++r) acc[r] = sMQ[wv][(r + half * 8) * 32 + col];
#pragma unroll
    for (int s = 0; s < 64; ++s) {
      // K slots kk0=4s+2*half, kk0+1 ; pair p = 2s+half
      v2f aF = *(const v2f*)(&sWcg2[(2 * s + half) * 32 + m * 2]);
      float ma = mb[s >> 2];
      v2f bF; bF.x = ma * rb0[s & 3]; bF.y = ma * rb1[s & 3];
      acc = wmma4(aF, bF, acc);
    }
#pragma unroll
    for (int r = 0; r < 8; ++r) sMQ[wv][(r + half * 8) * 32 + col] = acc[r];
  }
  __syncthreads();

  // featsOutT = (midT+quadT) @ W_prod + featsInT @ W_skip[z]
  int zz = spec[n];
  const float* Wsk = WskipS + (size_t)zz * 1024;  // swizzled per-species
  const float* fin = featsIn + (size_t)n * 512 + m * 32;
#pragma unroll
  for (int ct = 0; ct < 2; ++ct) {
    int col = ct * 16 + m;
    v8f acc = {0.f, 0.f, 0.f, 0.f, 0.f, 0.f, 0.f, 0.f};
#pragma unroll
    for (int s = 0; s < 8; ++s) {
      v2f a = *(const v2f*)(&sMQ[wv][m * 32 + 4 * s + aoff]);
      v2f b = *(const v2f*)(&sWprod[(2 * s + half) * 64 + col * 2]);
      acc = wmma4(a, b, acc);
    }
#pragma unroll
    for (int s = 0; s < 8; ++s) {
      v2f a = *(const v2f*)(fin + 4 * s + aoff);
      v2f b = *(const v2f*)(Wsk + (2 * s + half) * 64 + col * 2);
      acc = wmma4(a, b, acc);
    }
    float* fo = featsOut + (size_t)n * 512 + col;
#pragma unroll
    for (int r = 0; r < 8; ++r) fo[(r + half * 8) * 32] = acc[r];
    if (half == 0)  // invariant slice feats[:,:,0] = row M=0 (VGPR0, lanes 0-15)
      out[(size_t)n * (NLAYER * KCH) + layer * KCH + col] = acc[0];
  }
}

// ---------------------------------------------------------------- launcher
extern "C" void kernel_launch(void* const* d_in, const int* in_sizes, int n_in,
                              void* d_out, int out_size, void* d_ws, size_t ws_size,
                              hipStream_t stream) {
  const float* pos    = (const float*)d_in[0];
  const float* shifts = (const float*)d_in[1];
  const float* attrs  = (const float*)d_in[2];
  const float* Wemb   = (const float*)d_in[3];
  const float* Wup    = (const float*)d_in[4];
  const float* Wr1    = (const float*)d_in[5];
  const float* Wr2    = (const float*)d_in[6];
  const float* Wlin   = (const float*)d_in[7];
  const float* Wskip  = (const float*)d_in[8];
  const float* Wprod  = (const float*)d_in[9];
  const float* Wcg    = (const float*)d_in[10];
  const float* Wcg2   = (const float*)d_in[11];
  const int*   eidx   = (const int*)d_in[12];
  int N = in_sizes[0] / 3;
  int E = in_sizes[1] / 3;
  float* out = (float*)d_out;
  (void)n_in; (void)out_size; (void)ws_size;

  char* ws = (char*)d_ws;
  size_t off = 0;
  auto carve = [&](size_t bytes) -> char* {
    char* p = ws + off;
    off += (bytes + 255) & ~(size_t)255;
    return p;
  };
  float* Yb     = (float*)carve((size_t)E * 16 * sizeof(float));
  float* EFb    = (float*)carve((size_t)E * 8 * sizeof(float));
  float* fA     = (float*)carve((size_t)N * 512 * sizeof(float));
  float* fB     = (float*)carve((size_t)N * 512 * sizeof(float));
  float* upb    = (float*)carve((size_t)N * 512 * sizeof(float));
  float* aggb   = (float*)carve((size_t)N * 512 * sizeof(float));
  int*   spec   = (int*)carve((size_t)N * sizeof(int));
  float* WupS   = (float*)carve((size_t)NLAYER * 1024 * sizeof(float));
  float* WlinS  = (float*)carve((size_t)NLAYER * 1024 * sizeof(float));
  float* WprodS = (float*)carve((size_t)NLAYER * 1024 * sizeof(float));
  float* WskipS = (float*)carve((size_t)NLAYER * ZDIM * 1024 * sizeof(float));
  float* Wcg2S  = (float*)carve((size_t)4096 * sizeof(float));

  // pre-swizzle all weight matrices into fragment pair layout
  swizzle_w_kernel<<<(NLAYER * 1024 + 255) / 256, 256, 0, stream>>>(Wup, WupS, NLAYER);
  swizzle_w_kernel<<<(NLAYER * 1024 + 255) / 256, 256, 0, stream>>>(Wlin, WlinS, NLAYER);
  swizzle_w_kernel<<<(NLAYER * 1024 + 255) / 256, 256, 0, stream>>>(Wprod, WprodS, NLAYER);
  swizzle_w_kernel<<<(NLAYER * ZDIM * 1024 + 255) / 256, 256, 0, stream>>>(Wskip, WskipS, NLAYER * ZDIM);
  swizzle_w16_kernel<<<(4096 + 255) / 256, 256, 0, stream>>>(Wcg2, Wcg2S);

  geom_kernel<<<(E + 255) / 256, 256, 0, stream>>>(pos, shifts, eidx, Yb, EFb, E);
  embed_kernel<<<(N * KCH + 255) / 256, 256, 0, stream>>>(attrs, Wemb, fA, spec, N);

  float* fin = fA;
  float* fout = fB;
  for (int layer = 0; layer < NLAYER; ++layer) {
    zero_kernel<<<(N * 512 + 255) / 256, 256, 0, stream>>>(aggb, N * 512);
    linear_up_kernel<<<N / 8, 256, 0, stream>>>(fin, WupS + layer * 1024, upb);
    edge_kernel<<<E / 8, 256, 0, stream>>>(Yb, EFb, Wr1 + layer * 256,
                                           Wr2 + layer * 1024, Wcg, upb, eidx,
                                           aggb, E);
    node_update_kernel<<<N / 8, 256, 0, stream>>>(
        aggb, fin, WlinS + layer * 1024, WprodS + layer * 1024,
        WskipS + (size_t)layer * ZDIM * 1024, Wcg2S, spec, fout, out, layer);
    float* t = fin; fin = fout; fout = t;
  }
}